// LearnableVQ_9723805958205
// MI455X (gfx1250) — compile-verified
//
#include <hip/hip_runtime.h>
#include <hip/hip_bf16.h>

typedef __attribute__((ext_vector_type(2))) float v2f;
typedef __attribute__((ext_vector_type(8))) float v8f;
typedef __attribute__((ext_vector_type(4))) unsigned int v4u;
typedef __attribute__((ext_vector_type(8))) int v8i_;
typedef __attribute__((ext_vector_type(4))) int v4i;

#define EPSC   0.01f
#define GAMMAC 0.99f
#define NB 4
#define NH 8
#define NS 4096
#define ND 64
#define NL 512

#define CB_STRIDE 68   // padded LDS row stride (floats): conflict-free, 16B-aligned rows

// ---- output layout (floats) ----
#define VHAT_OFF   0                                   // B*H*S*D = 8388608
#define Z_OFF      (NB*NH*NS*ND)                       // 8388608, size B*H*S
#define COMMIT_OFF (NB*NH*NS*ND + NB*NH*NS)            // 8519680
#define LCB_OFF    (COMMIT_OFF + 1)                    // 8519681

// ---- workspace layout (floats) ----
#define WS_CODEBOOK 0                                  // H*L*D
#define WS_C2       (NH*NL*ND)                         // H*L
#define WS_CSUMHAT  (NH*NL*ND + NH*NL)                 // H*L*D
#define WS_CCNTHAT  (2*NH*NL*ND + NH*NL)               // H*L
#define WS_COMMIT   (2*NH*NL*ND + 2*NH*NL)             // 1

// ============================================================================
// TDM: issue one tensor_load_to_lds moving a 16x64 f32 tile (row-major,
// tensor_dim0_stride = 64) into LDS with 4-DWORD padding every 64 DWORDs
// (=> LDS row stride 68 floats). Descriptor per CDNA5 ISA.
// This toolchain's builtin is the 6-arg clang-23 form.
// ============================================================================
__device__ __forceinline__ void tdm_load_tile(const float* gsrc, unsigned lds_byte_off) {
  const unsigned long long ga = (unsigned long long)(uintptr_t)gsrc;
  v4u g0;
  g0.x = 1u;                                        // count=1 (valid user D#)
  g0.y = lds_byte_off;                              // lds_addr (bytes)
  g0.z = (unsigned)(ga & 0xFFFFFFFFu);              // global_addr[31:0]
  g0.w = (unsigned)((ga >> 32) & 0x1FFFFFFu)        // global_addr[56:32]
         | (2u << 30);                              // type = 2 ("image")
  v8i_ g1;
  g1[0] = (int)((2u << 16)    // data_size = 4 bytes
              | (1u << 20)    // pad_enable
              | (5u << 22)    // pad_interval: 64 DWORDs
              | (3u << 25));  // pad_amount:   4 DWORDs
  g1[1] = (int)(64u << 16);   // [47:32] atomic_bar_addr=0 | tensor_dim0 = 64 (lo16)
  g1[2] = (int)(16u << 16);   // tensor_dim0 hi=0 | tensor_dim1 = 16 (lo16)
  g1[3] = (int)(64u << 16);   // tensor_dim1 hi=0 | tile_dim0 = 64
  g1[4] = 16;                 // tile_dim1 = 16, tile_dim2 = 0
  g1[5] = 64;                 // tensor_dim0_stride = 64 (lo32)
  g1[6] = 0;                  // stride hi / tensor_dim1_stride lo (unused, 2D)
  g1[7] = 0;
  const v4i  g2 = {0, 0, 0, 0};
  const v4i  g3 = {0, 0, 0, 0};
  const v8i_ g4 = {0, 0, 0, 0, 0, 0, 0, 0};
  __builtin_amdgcn_tensor_load_to_lds(g0, g1, g2, g3, g4, 0);
}

// ============================================================================
// Kernel 1: codebook = c_sum / clip(c_count, EPS);  c2 = ||codebook||^2
//           zero the scatter accumulators and output scalar slots.
// grid = H*L blocks, 64 threads
// ============================================================================
__global__ __launch_bounds__(64) void vq_prep(const float* __restrict__ c_sum,
                                              const float* __restrict__ c_count,
                                              float* __restrict__ codebook,
                                              float* __restrict__ c2,
                                              float* __restrict__ c_sum_hat,
                                              float* __restrict__ c_count_hat,
                                              float* __restrict__ commit_acc,
                                              float* __restrict__ out) {
  const int row = blockIdx.x;            // h*L + l
  const int t   = threadIdx.x;           // 0..63 (D)
  const float denom = fmaxf(c_count[row], EPSC);
  const float cb = c_sum[(size_t)row * ND + t] / denom;
  codebook[(size_t)row * ND + t]  = cb;
  c_sum_hat[(size_t)row * ND + t] = 0.0f;

  float p = cb * cb;
  for (int off = 16; off >= 1; off >>= 1) p += __shfl_xor(p, off, 32);
  __shared__ float s[2];
  if ((t & 31) == 0) s[t >> 5] = p;
  __syncthreads();
  if (t == 0) {
    c2[row]          = s[0] + s[1];
    c_count_hat[row] = 0.0f;
  }
  if (row == 0 && t == 0) {
    *commit_acc     = 0.0f;
    out[COMMIT_OFF] = 0.0f;
    out[LCB_OFF]    = 0.0f;
  }
}

// ============================================================================
// Kernel 2: TDM double-buffered codebook staging + V_WMMA_F32_16X16X4_F32
//           distances, argmin over L, vecs_hat gather, EMA scatter, commit loss.
// grid = B*H*(S/128) blocks, 256 threads (8 wave32, one 16-row M-tile each)
// ============================================================================
__global__ __launch_bounds__(256) void vq_main(const float* __restrict__ vecs,
                                               const float* __restrict__ loss_mask,
                                               const float* __restrict__ codebook,
                                               const float* __restrict__ c2,
                                               float* __restrict__ c_sum_hat,
                                               float* __restrict__ c_count_hat,
                                               float* __restrict__ commit_acc,
                                               float* __restrict__ out_vhat,
                                               float* __restrict__ out_z) {
  __shared__ float s_cb[2][16 * CB_STRIDE];   // double-buffered padded 16x64 tiles
  __shared__ float s_c2[2][16];

  const int lane = threadIdx.x & 31;
  const int wave = threadIdx.x >> 5;
  const int chunks = NS / 128;
  const int bh    = blockIdx.x / chunks;      // b*H + h
  const int chunk = blockIdx.x % chunks;
  const int b = bh / NH;
  const int h = bh % NH;
  const int s0 = chunk * 128 + wave * 16;     // first of this wave's 16 rows

  const int m16 = lane & 15;                  // row of A tile / col of B tile
  const int kh  = lane >> 4;                  // K half (ISA 16x4 fp32 A layout)

  const float* cbh = codebook + (size_t)h * NL * ND;
  const float* c2h = c2 + h * NL;

  // ---- preload A fragments: lane holds half of vecs row m = s0+m16 ----
  const float* vrow = vecs + (((size_t)bh) * NS + s0 + m16) * ND;
  v2f afrag[16];
  float partial = 0.0f;
#pragma unroll
  for (int k0 = 0; k0 < 16; ++k0) {
    afrag[k0] = *(const v2f*)(vrow + k0 * 4 + 2 * kh);
    partial += afrag[k0].x * afrag[k0].x + afrag[k0].y * afrag[k0].y;
  }
  const float v2row = partial + __shfl_xor(partial, 16, 32);

  float minval[8];
  int   minidx[8];
#pragma unroll
  for (int r = 0; r < 8; ++r) { minval[r] = 3.4e38f; minidx[r] = 0; }

  // ---- prologue: wave 0 launches TDM for tile 0, stages c2 row 0 ----
  if (wave == 0) {
    if (lane == 0)
      tdm_load_tile(cbh, (unsigned)(uintptr_t)&s_cb[0][0]);
    if (lane < 16) s_c2[0][lane] = c2h[lane];
  }

  // ---- sweep the 32 codebook N-tiles (TDM double-buffered) ----
  for (int tile = 0; tile < NL / 16; ++tile) {
    const int buf = tile & 1;
    if (wave == 0) __builtin_amdgcn_s_wait_tensorcnt(0);  // tile's TDM complete
    __syncthreads();                                      // visible to all waves
    if (tile + 1 < NL / 16 && wave == 0) {                // prefetch next tile
      if (lane == 0)
        tdm_load_tile(cbh + (size_t)(tile + 1) * 16 * ND,
                      (unsigned)(uintptr_t)&s_cb[buf ^ 1][0]);
      if (lane < 16) s_c2[buf ^ 1][lane] = c2h[(tile + 1) * 16 + lane];
    }

    v8f acc = {0.f, 0.f, 0.f, 0.f, 0.f, 0.f, 0.f, 0.f};
#pragma unroll
    for (int k0 = 0; k0 < 16; ++k0) {
      // B fragment: B[k][n] = codebook[n][k], n = lane%16, K half = kh
      const v2f bfrag = *(const v2f*)(&s_cb[buf][m16 * CB_STRIDE + k0 * 4 + 2 * kh]);
      acc = __builtin_amdgcn_wmma_f32_16x16x4_f32(
          /*neg_a=*/false, afrag[k0], /*neg_b=*/false, bfrag,
          /*c_mod=*/(short)0, acc, /*reuse_a=*/false, /*reuse_b=*/false);
    }

    // score(m, n) = |c_n|^2 - 2 * <v_m, c_n>   (add |v_m|^2 at the very end)
    const float c2v = s_c2[buf][m16];
    const int   n   = tile * 16 + m16;
#pragma unroll
    for (int r = 0; r < 8; ++r) {
      const float score = c2v - 2.0f * acc[r];
      if (score < minval[r]) { minval[r] = score; minidx[r] = n; }
    }
  }

  // ---- argmin across the 16 lanes sharing each output row ----
#pragma unroll
  for (int r = 0; r < 8; ++r) {
#pragma unroll
    for (int off = 8; off >= 1; off >>= 1) {
      const float ov = __shfl_xor(minval[r], off, 32);
      const int   oi = __shfl_xor(minidx[r], off, 32);
      if (ov < minval[r] || (ov == minval[r] && oi < minidx[r])) {
        minval[r] = ov; minidx[r] = oi;
      }
    }
  }
  // lanes 0..15 hold result for m=r, lanes 16..31 for m=r+8 (replicated)

  const size_t zbase = ((size_t)bh) * NS + s0;

  // ---- z output + commit-loss contribution ----
#pragma unroll
  for (int r = 0; r < 8; ++r) {
    const float v2m = __shfl(v2row, r + 8 * kh, 32);   // |v|^2 of row m=r+8*kh
    if (m16 == 0) {
      const int m = r + 8 * kh;
      out_z[zbase + m] = (float)minidx[r];
      const float mask = loss_mask[(size_t)b * NS + s0 + m];
      atomicAdd(commit_acc, mask * (minval[r] + v2m));
    }
  }

  // ---- vecs_hat gather + EMA scatter ----
#pragma unroll
  for (int m = 0; m < 16; ++m) {
    const int idxm = __shfl(minidx[m & 7], (m < 8) ? 0 : 16, 32);
    const float* crow = cbh + (size_t)idxm * ND;
    float* orow = out_vhat + (zbase + m) * ND;
    const v2f cv = *(const v2f*)(crow + lane * 2);
    *(v2f*)(orow + lane * 2) = cv;

    const float mask = loss_mask[(size_t)b * NS + s0 + m];
    if (mask != 0.0f) {
      const v2f vv = *(const v2f*)(vecs + (zbase + m) * ND + lane * 2);
      float* dst = c_sum_hat + ((size_t)h * NL + idxm) * ND + lane * 2;
      atomicAdd(dst + 0, mask * vv.x);
      atomicAdd(dst + 1, mask * vv.y);
      if (lane == 0) atomicAdd(c_count_hat + h * NL + idxm, mask);
    }
  }
}

// ============================================================================
// Kernel 3: l_codebook = gamma^2*(||c_sum-c_sum_hat||^2 + ||c_count-c_count_hat||^2)
//           l_commit  = commit_acc / (B*S)
// ============================================================================
__global__ __launch_bounds__(256) void vq_loss(const float* __restrict__ c_sum,
                                               const float* __restrict__ c_count,
                                               const float* __restrict__ c_sum_hat,
                                               const float* __restrict__ c_count_hat,
                                               const float* __restrict__ commit_acc,
                                               float* __restrict__ out) {
  const size_t NSUM = (size_t)NH * NL * ND;
  const size_t NCNT = (size_t)NH * NL;
  const size_t i0     = blockIdx.x * (size_t)blockDim.x + threadIdx.x;
  const size_t stride = gridDim.x * (size_t)blockDim.x;

  float local = 0.0f;
  for (size_t j = i0; j < NSUM; j += stride) {
    const float d = c_sum[j] - c_sum_hat[j];
    local += d * d;
  }
  for (size_t j = i0; j < NCNT; j += stride) {
    const float d = c_count[j] - c_count_hat[j];
    local += d * d;
  }
  for (int off = 16; off >= 1; off >>= 1) local += __shfl_xor(local, off, 32);

  __shared__ float s[8];
  const int lane = threadIdx.x & 31, w = threadIdx.x >> 5;
  if (lane == 0) s[w] = local;
  __syncthreads();
  if (threadIdx.x == 0) {
    float sum = 0.0f;
    for (int k = 0; k < 8; ++k) sum += s[k];
    atomicAdd(out + LCB_OFF, GAMMAC * GAMMAC * sum);
  }
  if (blockIdx.x == 0 && threadIdx.x == 0) {
    out[COMMIT_OFF] = (*commit_acc) * (1.0f / ((float)NB * (float)NS));
  }
}

// ============================================================================
extern "C" void kernel_launch(void* const* d_in, const int* in_sizes, int n_in,
                              void* d_out, int out_size, void* d_ws, size_t ws_size,
                              hipStream_t stream) {
  const float* vecs      = (const float*)d_in[0];   // (B,H,S,D)
  const float* loss_mask = (const float*)d_in[1];   // (B,S)
  const float* c_sum     = (const float*)d_in[2];   // (H,L,D)
  const float* c_count   = (const float*)d_in[3];   // (H,L)

  float* ws = (float*)d_ws;
  float* codebook    = ws + WS_CODEBOOK;
  float* c2          = ws + WS_C2;
  float* c_sum_hat   = ws + WS_CSUMHAT;
  float* c_count_hat = ws + WS_CCNTHAT;
  float* commit_acc  = ws + WS_COMMIT;

  float* out = (float*)d_out;

  vq_prep<<<NH * NL, 64, 0, stream>>>(c_sum, c_count, codebook, c2,
                                      c_sum_hat, c_count_hat, commit_acc, out);

  vq_main<<<NB * NH * (NS / 128), 256, 0, stream>>>(
      vecs, loss_mask, codebook, c2, c_sum_hat, c_count_hat, commit_acc,
      out + VHAT_OFF, out + Z_OFF);

  vq_loss<<<256, 256, 0, stream>>>(c_sum, c_count, c_sum_hat, c_count_hat,
                                   commit_acc, out);
}